// InterAgg_64020782514705
// MI455X (gfx1250) — compile-verified
//
#include <hip/hip_runtime.h>

#define B_SZ   4096
#define DEG    32
#define P_SZ   2048
#define FEAT   64
#define KSEL   16
#define EPSV   1e-8f
#define LDA    68   // 68 mod 64 == 4 -> conflict-free A-fragment reads, 16B-aligned rows

typedef __attribute__((ext_vector_type(2))) float v2f;
typedef __attribute__((ext_vector_type(8))) float v8f;

// ---- CDNA5 async global->LDS (writes LDS directly from the memory pipe, ASYNCcnt-tracked) ----
__device__ __forceinline__ void async_ld_b128(const void* lds_dst, const void* gsrc)
{
    const unsigned lds_off = (unsigned)(size_t)lds_dst;   // low 32 bits of generic ptr = LDS byte offset
    asm volatile("global_load_async_to_lds_b128 %0, %1, off"
                 :: "v"(lds_off), "v"(gsrc) : "memory");
}
__device__ __forceinline__ void wait_async0()
{
    asm volatile("s_wait_asynccnt 0x0" ::: "memory");
}

// One 16x16 output tile of A(16xK=64) @ B(64x16) using V_WMMA_F32_16X16X4_F32.
// A layout: lane(0-15)=M, VGPR0=K(+0/+2 by half), VGPR1=K(+1/+3). B mirrors with N on lanes.
__device__ __forceinline__ v8f wmma_k64(const float* __restrict__ A, int lda,
                                        const float* __restrict__ Bm, int ldb,
                                        int row0, int col0, v8f acc, int lane)
{
    const int hi  = lane >> 4;
    const int idx = lane & 15;
    #pragma unroll
    for (int k = 0; k < 64; k += 4) {
        const int ka = k + 2 * hi;
        v2f a, b;
        a.x = A[(row0 + idx) * lda + ka];
        a.y = A[(row0 + idx) * lda + ka + 1];
        b.x = Bm[ka * ldb + col0 + idx];
        b.y = Bm[(ka + 1) * ldb + col0 + idx];
        acc = __builtin_amdgcn_wmma_f32_16x16x4_f32(false, a, false, b, (short)0, acc,
                                                    false, false);
    }
    return acc;
}

// xor-reduce within each 16-lane half of a wave32; every lane of a half ends with the half-sum
__device__ __forceinline__ float red16(float x)
{
    x += __shfl_xor(x, 1, 32);
    x += __shfl_xor(x, 2, 32);
    x += __shfl_xor(x, 4, 32);
    x += __shfl_xor(x, 8, 32);
    return x;
}

__global__ void init_kernel(float* pos_sum, float* loss_sum)
{
    if (threadIdx.x < 3) { pos_sum[threadIdx.x] = 0.f; loss_sum[threadIdx.x] = 0.f; }
}

// mean-of-cos(pos_emb_r, relu(features[pos_nodes] @ R)) accumulation (numerator Σ only)
__global__ void pos_kernel(const int* __restrict__ pos_nodes,
                           const float* __restrict__ features,
                           const float* __restrict__ rsim,
                           const float* __restrict__ pos_embs,
                           float* pos_sum)
{
    __shared__ float Rlds[64 * 64];
    __shared__ float Alds[32 * LDA];
    __shared__ float nrm_acc[32];
    __shared__ float num_acc[3][32];
    __shared__ int   idxs[32];

    const int t = threadIdx.x;
    if (t < 32) {
        idxs[t] = pos_nodes[blockIdx.x * 32 + t];
        nrm_acc[t] = 0.f;
        num_acc[0][t] = 0.f; num_acc[1][t] = 0.f; num_acc[2][t] = 0.f;
    }
    // stream rsimTrans into LDS asynchronously; overlaps with index staging below
    for (int q = t; q < 1024; q += 256)
        async_ld_b128(&Rlds[q * 4], (const char*)rsim + (size_t)q * 16);
    __syncthreads();                                   // publishes idxs

    for (int q = t; q < 512; q += 256) {
        const int row = q >> 4, seg = q & 15;
        async_ld_b128(&Alds[row * LDA + seg * 4],
                      (const char*)features + ((size_t)idxs[row] * 64 + seg * 4) * 4);
    }
    wait_async0();
    __syncthreads();

    const int w = t >> 5, lane = t & 31;
    const int mt = w >> 2, nt = w & 3;
    const int hi = lane >> 4, idx = lane & 15;
    v8f acc = {0.f,0.f,0.f,0.f,0.f,0.f,0.f,0.f};
    acc = wmma_k64(Alds, LDA, Rlds, 64, mt * 16, nt * 16, acc, lane);

    const float pe0 = pos_embs[0 * 64 + nt * 16 + idx];
    const float pe1 = pos_embs[1 * 64 + nt * 16 + idx];
    const float pe2 = pos_embs[2 * 64 + nt * 16 + idx];
    #pragma unroll
    for (int v = 0; v < 8; ++v) {
        const float val = fmaxf(acc[v], 0.f);
        float ns = red16(val * val);
        float n0 = red16(val * pe0);
        float n1 = red16(val * pe1);
        float n2 = red16(val * pe2);
        if (idx == 0) {
            const int row = mt * 16 + v + hi * 8;
            atomicAdd(&nrm_acc[row], ns);
            atomicAdd(&num_acc[0][row], n0);
            atomicAdd(&num_acc[1][row], n1);
            atomicAdd(&num_acc[2][row], n2);
        }
    }
    __syncthreads();

    if (t < 32) {
        const float nb = fmaxf(sqrtf(nrm_acc[t]), EPSV);
        #pragma unroll
        for (int r = 0; r < 3; ++r) {
            float s2 = 0.f;
            for (int j = 0; j < 64; ++j) { float e = pos_embs[r * 64 + j]; s2 += e * e; }
            const float pen = fmaxf(sqrtf(s2), EPSV);
            atomicAdd(&pos_sum[r], num_acc[r][t] / (pen * nb));
        }
    }
}

// one block per (batch row, relation): gather 32 neighbors, WMMA n_sim tiles, cos scores,
// stable top-16, aggregate + relu(agg@W_r) into cat slice, accumulate loss
__global__ void relation_kernel(const int* __restrict__ ng1, const int* __restrict__ ng2,
                                const int* __restrict__ ng3,
                                const float* __restrict__ features,
                                const float* __restrict__ rsim,
                                const float* __restrict__ pos_embs,
                                const float* __restrict__ W1, const float* __restrict__ W2,
                                const float* __restrict__ W3,
                                float* __restrict__ cat,
                                const float* __restrict__ pos_sum,
                                float* loss_sum)
{
    __shared__ float Rlds[64 * 64];
    __shared__ float Alds[32 * LDA];
    __shared__ float num_acc[32], nrm_acc[32], scores[32], sqv[32], agg[64];
    __shared__ int   sel[32], idxs[32];

    const int b = blockIdx.x;
    const int r = blockIdx.y;
    const int* neigh = (r == 0) ? ng1 : (r == 1) ? ng2 : ng3;
    const float* W   = (r == 0) ? W1  : (r == 1) ? W2  : W3;

    const int t = threadIdx.x;
    if (t < 32) {
        idxs[t] = neigh[b * DEG + t];
        num_acc[t] = 0.f;
        nrm_acc[t] = 0.f;
    }
    for (int q = t; q < 1024; q += 256)
        async_ld_b128(&Rlds[q * 4], (const char*)rsim + (size_t)q * 16);
    __syncthreads();                                   // publishes idxs

    for (int q = t; q < 512; q += 256) {
        const int row = q >> 4, seg = q & 15;
        async_ld_b128(&Alds[row * LDA + seg * 4],
                      (const char*)features + ((size_t)idxs[row] * 64 + seg * 4) * 4);
    }
    wait_async0();
    __syncthreads();

    const int w = t >> 5, lane = t & 31;
    const int mt = w >> 2, nt = w & 3;
    const int hi = lane >> 4, idx = lane & 15;
    v8f acc = {0.f,0.f,0.f,0.f,0.f,0.f,0.f,0.f};
    acc = wmma_k64(Alds, LDA, Rlds, 64, mt * 16, nt * 16, acc, lane);

    const float pe = pos_embs[r * 64 + nt * 16 + idx];
    #pragma unroll
    for (int v = 0; v < 8; ++v) {
        const float val = fmaxf(acc[v], 0.f);   // relu(n_sim)
        float nm = red16(val * pe);
        float ns = red16(val * val);
        if (idx == 0) {
            const int row = mt * 16 + v + hi * 8;
            atomicAdd(&num_acc[row], nm);
            atomicAdd(&nrm_acc[row], ns);
        }
    }
    __syncthreads();

    if (t < 32) {
        float s2 = 0.f;
        for (int j = 0; j < 64; ++j) { float e = pos_embs[r * 64 + j]; s2 += e * e; }
        const float pen = fmaxf(sqrtf(s2), EPSV);
        const float nb  = fmaxf(sqrtf(nrm_acc[t]), EPSV);
        scores[t] = num_acc[t] / (pen * nb);
    }
    __syncthreads();

    if (t < 32) {
        const float sd = scores[t];
        int rank = 0;
        for (int j = 0; j < 32; ++j) {
            const float sj = scores[j];
            rank += (sj > sd) || (sj == sd && j < t);
        }
        const int keep = (rank < KSEL);
        sel[t] = keep;
        const float pmean = pos_sum[r] * (1.f / (float)P_SZ);
        const float d = sd - pmean;
        sqv[t] = keep ? d * d : 0.f;
    }
    __syncthreads();

    if (t == 0) {
        float ls = 0.f;
        for (int j = 0; j < 32; ++j) ls += sqv[j];
        atomicAdd(&loss_sum[r], ls);
    }
    if (t < 64) {
        float a = 0.f;
        for (int d = 0; d < 32; ++d)
            if (sel[d]) a += Alds[d * LDA + t];
        agg[t] = a * (1.f / (float)KSEL);
    }
    __syncthreads();

    if (t < 64) {
        float o = 0.f;
        for (int f = 0; f < 64; ++f) o += agg[f] * W[f * 64 + t];
        cat[(size_t)b * 256 + 64 + r * 64 + t] = fmaxf(o, 0.f);
    }
}

__global__ void center_kernel(const int* __restrict__ nodes,
                              const float* __restrict__ features,
                              float* __restrict__ cat)
{
    const int gid = blockIdx.x * blockDim.x + threadIdx.x;   // B*16 float4 slots
    const int b = gid >> 4, seg = gid & 15;
    const float4* F4 = (const float4*)features;
    float4* C4 = (float4*)cat;
    C4[(size_t)b * 64 + seg] = F4[(size_t)nodes[b] * 16 + seg];
}

// out[64,4096] = relu(cat[4096,256] @ weight[256,64]).T ; block0 also writes the loss scalar
__global__ void gemm_kernel(const float* __restrict__ cat,
                            const float* __restrict__ weight,
                            float* __restrict__ out,
                            const float* __restrict__ loss_sum)
{
    __shared__ float Alds[32 * LDA];
    __shared__ float Blds[64 * 64];

    const int t = threadIdx.x;
    const int m0 = blockIdx.x * 32;
    const int w = t >> 5, lane = t & 31;
    const int mt = w >> 2, nt = w & 3;
    const int hi = lane >> 4, idx = lane & 15;

    v8f acc = {0.f,0.f,0.f,0.f,0.f,0.f,0.f,0.f};

    for (int kc = 0; kc < 4; ++kc) {
        for (int q = t; q < 512; q += 256) {
            const int row = q >> 4, seg = q & 15;
            async_ld_b128(&Alds[row * LDA + seg * 4],
                          (const char*)cat + ((size_t)(m0 + row) * 256 + kc * 64 + seg * 4) * 4);
        }
        for (int q = t; q < 1024; q += 256)
            async_ld_b128(&Blds[q * 4], (const char*)weight + ((size_t)kc * 4096 + q * 4) * 4);
        wait_async0();
        __syncthreads();
        acc = wmma_k64(Alds, LDA, Blds, 64, mt * 16, nt * 16, acc, lane);
        __syncthreads();
    }

    #pragma unroll
    for (int v = 0; v < 8; ++v) {
        const int row = m0 + mt * 16 + v + hi * 8;
        const int col = nt * 16 + idx;
        out[(size_t)col * B_SZ + row] = fmaxf(acc[v], 0.f);
    }
    if (blockIdx.x == 0 && t == 0) {
        out[(size_t)64 * B_SZ] =
            (loss_sum[0] + loss_sum[1] + loss_sum[2]) * (1.f / (3.f * B_SZ * KSEL));
    }
}

extern "C" void kernel_launch(void* const* d_in, const int* in_sizes, int n_in,
                              void* d_out, int out_size, void* d_ws, size_t ws_size,
                              hipStream_t stream)
{
    const int*   nodes     = (const int*)d_in[0];
    const int*   neigh1    = (const int*)d_in[2];
    const int*   neigh2    = (const int*)d_in[3];
    const int*   neigh3    = (const int*)d_in[4];
    const int*   pos_nodes = (const int*)d_in[5];
    const float* features  = (const float*)d_in[6];
    const float* weight    = (const float*)d_in[7];
    const float* rsim      = (const float*)d_in[8];
    const float* pos_embs  = (const float*)d_in[9];
    const float* W1        = (const float*)d_in[10];
    const float* W2        = (const float*)d_in[11];
    const float* W3        = (const float*)d_in[12];

    float* ws       = (float*)d_ws;
    float* cat      = ws;                       // [4096, 256]
    float* pos_sum  = ws + (size_t)B_SZ * 256;  // [3]
    float* loss_sum = pos_sum + 3;              // [3]
    float* out      = (float*)d_out;            // [64*4096] ++ [1]

    init_kernel<<<1, 64, 0, stream>>>(pos_sum, loss_sum);
    pos_kernel<<<P_SZ / 32, 256, 0, stream>>>(pos_nodes, features, rsim, pos_embs, pos_sum);
    relation_kernel<<<dim3(B_SZ, 3), 256, 0, stream>>>(neigh1, neigh2, neigh3, features,
                                                       rsim, pos_embs, W1, W2, W3,
                                                       cat, pos_sum, loss_sum);
    center_kernel<<<B_SZ * 16 / 256, 256, 0, stream>>>(nodes, features, cat);
    gemm_kernel<<<B_SZ / 32, 256, 0, stream>>>(cat, weight, out, loss_sum);
}